// RejectionSampler_14259291422831
// MI455X (gfx1250) — compile-verified
//
#include <hip/hip_runtime.h>
#include <stdint.h>

#define BB 64
#define KK 8
#define VV 32000
#define ROWS (BB * (KK + 1))   // 576
#define INVALID_TOKEN_ID (-1)
#define THREADS 256
#define NV4 (VV / 4)           // 8000 float4 per row
#define CHUNK 256              // float4 per stage per array (4KB)
#define NFULL (NV4 / CHUNK)    // 31 full double-buffered stages
#define TAIL (NV4 - NFULL * CHUNK)  // 64 float4 tail

// -------- CDNA5 async global->LDS copy (ASYNCcnt-tracked) ------------------
// the LDS address is taken from the VDST VGPR; low 32 bits of a generic LDS pointer are
// the LDS byte offset (ISA 10.2), so a truncating cast gives the VDST value.
__device__ __forceinline__ void async_copy_b128(const float4* gsrc, float4* ldst) {
    const uint32_t lds_off = (uint32_t)(uintptr_t)ldst;
    asm volatile("global_load_async_to_lds_b128 %0, %1, off"
                 :: "v"(lds_off), "v"(gsrc)
                 : "memory");
}
#define WAIT_ASYNCCNT(n) asm volatile("s_wait_asynccnt " #n ::: "memory")

// ---------------------------------------------------------------------------
// Kernel 1: per-(b,j) row, argmax over V of sample_prob / exp_noise.
//   j <  K : sample_prob ∝ max(target - draft, 1e-5)   (row-sum normalization
//            is argmax-invariant, so it is skipped -> single pass)
//   j == K : sample_prob = target[b,K,:]
// HBM-bound streaming: async global->LDS double buffering, LDS b128 reads,
// per-thread scan, LDS tree reduction with first-index tie-break (jnp.argmax).
// ---------------------------------------------------------------------------
__global__ __launch_bounds__(THREADS) void bonus_argmax_kernel(
    const float* __restrict__ draft_probs,    // [B,K,V]
    const float* __restrict__ target_probs,   // [B,K+1,V]
    const float* __restrict__ exp_noise,      // [B,K+1,V]
    int* __restrict__ bonus_ids)              // [B*(K+1)] in workspace
{
    const int row = blockIdx.x;         // 0 .. ROWS-1
    const int b   = row / (KK + 1);
    const int j   = row % (KK + 1);
    const int tid = threadIdx.x;

    const float4* __restrict__ t4 =
        (const float4*)(target_probs + ((size_t)b * (KK + 1) + j) * VV);
    const float4* __restrict__ n4 =
        (const float4*)(exp_noise    + ((size_t)b * (KK + 1) + j) * VV);

    __shared__ float4 sh_t[2][CHUNK];
    __shared__ float4 sh_n[2][CHUNK];
    __shared__ float4 sh_d[2][CHUNK];

    float bestVal = -1.0f;   // ratios are strictly positive
    int   bestIdx = 0;

    if (j < KK) {
        const float4* __restrict__ d4 =
            (const float4*)(draft_probs + ((size_t)b * KK + j) * VV);

        // prologue: stage 0 into buffer 0
        async_copy_b128(t4 + tid, &sh_t[0][tid]);
        async_copy_b128(n4 + tid, &sh_n[0][tid]);
        async_copy_b128(d4 + tid, &sh_d[0][tid]);

        for (int it = 0; it < NFULL; ++it) {
            const int buf = it & 1;
            if (it + 1 < NFULL) {
                const int nb  = buf ^ 1;
                const int off = (it + 1) * CHUNK + tid;
                async_copy_b128(t4 + off, &sh_t[nb][tid]);
                async_copy_b128(n4 + off, &sh_n[nb][tid]);
                async_copy_b128(d4 + off, &sh_d[nb][tid]);
                WAIT_ASYNCCNT(3);   // async loads complete in order: stage 'it' done
            } else {
                WAIT_ASYNCCNT(0);
            }
            __syncthreads();        // all waves' stage 'it' data visible in LDS

            const float4 t = sh_t[buf][tid];
            const float4 d = sh_d[buf][tid];
            const float4 n = sh_n[buf][tid];
            const float r0 = fmaxf(t.x - d.x, 1e-5f) / n.x;
            const float r1 = fmaxf(t.y - d.y, 1e-5f) / n.y;
            const float r2 = fmaxf(t.z - d.z, 1e-5f) / n.z;
            const float r3 = fmaxf(t.w - d.w, 1e-5f) / n.w;
            const int base = (it * CHUNK + tid) * 4;
            // per-thread indices strictly increase -> '>' keeps first max
            if (r0 > bestVal) { bestVal = r0; bestIdx = base + 0; }
            if (r1 > bestVal) { bestVal = r1; bestIdx = base + 1; }
            if (r2 > bestVal) { bestVal = r2; bestIdx = base + 2; }
            if (r3 > bestVal) { bestVal = r3; bestIdx = base + 3; }

            __syncthreads();        // reads done before buf is overwritten (stage it+2)
        }
        // tail: 64 float4, plain global loads (no masked async bookkeeping)
        if (tid < TAIL) {
            const int v = NFULL * CHUNK + tid;
            const float4 t = t4[v];
            const float4 d = d4[v];
            const float4 n = n4[v];
            const float r0 = fmaxf(t.x - d.x, 1e-5f) / n.x;
            const float r1 = fmaxf(t.y - d.y, 1e-5f) / n.y;
            const float r2 = fmaxf(t.z - d.z, 1e-5f) / n.z;
            const float r3 = fmaxf(t.w - d.w, 1e-5f) / n.w;
            const int base = v * 4;
            if (r0 > bestVal) { bestVal = r0; bestIdx = base + 0; }
            if (r1 > bestVal) { bestVal = r1; bestIdx = base + 1; }
            if (r2 > bestVal) { bestVal = r2; bestIdx = base + 2; }
            if (r3 > bestVal) { bestVal = r3; bestIdx = base + 3; }
        }
    } else {
        // prologue: stage 0
        async_copy_b128(t4 + tid, &sh_t[0][tid]);
        async_copy_b128(n4 + tid, &sh_n[0][tid]);

        for (int it = 0; it < NFULL; ++it) {
            const int buf = it & 1;
            if (it + 1 < NFULL) {
                const int nb  = buf ^ 1;
                const int off = (it + 1) * CHUNK + tid;
                async_copy_b128(t4 + off, &sh_t[nb][tid]);
                async_copy_b128(n4 + off, &sh_n[nb][tid]);
                WAIT_ASYNCCNT(2);
            } else {
                WAIT_ASYNCCNT(0);
            }
            __syncthreads();

            const float4 t = sh_t[buf][tid];
            const float4 n = sh_n[buf][tid];
            const float r0 = t.x / n.x;
            const float r1 = t.y / n.y;
            const float r2 = t.z / n.z;
            const float r3 = t.w / n.w;
            const int base = (it * CHUNK + tid) * 4;
            if (r0 > bestVal) { bestVal = r0; bestIdx = base + 0; }
            if (r1 > bestVal) { bestVal = r1; bestIdx = base + 1; }
            if (r2 > bestVal) { bestVal = r2; bestIdx = base + 2; }
            if (r3 > bestVal) { bestVal = r3; bestIdx = base + 3; }

            __syncthreads();
        }
        if (tid < TAIL) {
            const int v = NFULL * CHUNK + tid;
            const float4 t = t4[v];
            const float4 n = n4[v];
            const float r0 = t.x / n.x;
            const float r1 = t.y / n.y;
            const float r2 = t.z / n.z;
            const float r3 = t.w / n.w;
            const int base = v * 4;
            if (r0 > bestVal) { bestVal = r0; bestIdx = base + 0; }
            if (r1 > bestVal) { bestVal = r1; bestIdx = base + 1; }
            if (r2 > bestVal) { bestVal = r2; bestIdx = base + 2; }
            if (r3 > bestVal) { bestVal = r3; bestIdx = base + 3; }
        }
    }

    // Block-wide (max, first-index) reduction. Wave32-safe: plain LDS tree.
    __shared__ float s_val[THREADS];
    __shared__ int   s_idx[THREADS];
    s_val[tid] = bestVal;
    s_idx[tid] = bestIdx;
    __syncthreads();
    for (int s = THREADS / 2; s > 0; s >>= 1) {
        if (tid < s) {
            const float ov = s_val[tid + s];
            const int   oi = s_idx[tid + s];
            if (ov > s_val[tid] || (ov == s_val[tid] && oi < s_idx[tid])) {
                s_val[tid] = ov;
                s_idx[tid] = oi;
            }
        }
        __syncthreads();
    }
    if (tid == 0) bonus_ids[row] = s_idx[0];
}

// ---------------------------------------------------------------------------
// Kernel 2: accept/cumprod logic + output composition. One thread per batch
// row; a handful of gathered scalar loads.
// ---------------------------------------------------------------------------
__global__ __launch_bounds__(64) void finalize_kernel(
    const int*   __restrict__ draft_token_ids,  // [B,K]
    const float* __restrict__ draft_probs,      // [B,K,V]
    const float* __restrict__ target_probs,     // [B,K+1,V]
    const float* __restrict__ uniform_samples,  // [B,K]
    const int*   __restrict__ bonus_ids,        // [B*(K+1)]
    int* __restrict__ out)                      // [B,K+1]
{
    const int b = threadIdx.x;
    if (b >= BB) return;

    int outrow[KK + 1];
    int mask = 1;          // running cumprod of accepts
    int first_zero = -1;   // first rejected slot; -1 = none yet

    for (int k = 0; k < KK; ++k) {
        const int id = draft_token_ids[b * KK + k];
        const float pd = draft_probs[((size_t)b * KK + k) * VV + id];
        const float pt = target_probs[((size_t)b * (KK + 1) + k) * VV + id];
        const int acc = (uniform_samples[b * KK + k] <= pt / pd) ? 1 : 0;
        mask &= acc;
        if (mask == 0 && first_zero < 0) first_zero = k;
        outrow[k] = mask ? id : INVALID_TOKEN_ID;
    }
    outrow[KK] = INVALID_TOKEN_ID;
    if (first_zero < 0) first_zero = KK;   // all accepted -> bonus in slot K
    outrow[first_zero] = bonus_ids[b * (KK + 1) + first_zero];

    for (int jj = 0; jj <= KK; ++jj) out[b * (KK + 1) + jj] = outrow[jj];
}

// ---------------------------------------------------------------------------
// Launcher. Inputs in setup_inputs() order:
//   0: draft_token_ids [B,K] i32   1: draft_probs [B,K,V] f32
//   2: target_probs [B,K+1,V] f32  3: uniform_samples [B,K] f32
//   4: exp_noise [B,K+1,V] f32
// Output: [B,K+1] int32 token ids.
// ---------------------------------------------------------------------------
extern "C" void kernel_launch(void* const* d_in, const int* in_sizes, int n_in,
                              void* d_out, int out_size, void* d_ws, size_t ws_size,
                              hipStream_t stream) {
    (void)in_sizes; (void)n_in; (void)out_size; (void)ws_size;

    const int*   draft_token_ids = (const int*)  d_in[0];
    const float* draft_probs     = (const float*)d_in[1];
    const float* target_probs    = (const float*)d_in[2];
    const float* uniform_samples = (const float*)d_in[3];
    const float* exp_noise       = (const float*)d_in[4];

    int* out      = (int*)d_out;
    int* bonus_ws = (int*)d_ws;   // ROWS ints of scratch

    bonus_argmax_kernel<<<ROWS, THREADS, 0, stream>>>(
        draft_probs, target_probs, exp_noise, bonus_ws);

    finalize_kernel<<<1, 64, 0, stream>>>(
        draft_token_ids, draft_probs, target_probs, uniform_samples,
        bonus_ws, out);
}